// GARCH_86870008529312
// MI455X (gfx1250) — compile-verified
//
#include <hip/hip_runtime.h>
#include <cstdint>

// GARCH(1,1) volatility scan, MI455X (gfx1250), wave32.
// One wave per row:
//   TDM tensor_load_to_lds (hardware-padded layout) -> chunked parallel
//   linear-recurrence scan (Hillis-Steele over affine maps) -> sqrt in place
//   -> async store-from-LDS (coalesced 512B transactions).
// Memory-bound: 128 MB @ 23.3 TB/s ~= 5.5 us; critical path per row reduced
// from 4096 serial steps to ~300 via per-lane chunk decomposition.

#define T_LEN        4096
#define CHUNK        128                 // elements per lane
#define LDS_STRIDE   132                 // floats per chunk slot (128 + 4 pad)
#define ROWS_PER_BLK 2

typedef unsigned int u32;
typedef u32 v4u __attribute__((ext_vector_type(4)));
typedef int v4i __attribute__((ext_vector_type(4)));
typedef int v8i __attribute__((ext_vector_type(8)));

__global__ __launch_bounds__(32 * ROWS_PER_BLK)
void garch_scan_kernel(const float* __restrict__ x,
                       const float* __restrict__ omega_log,
                       const float* __restrict__ alpha_log,
                       const float* __restrict__ beta_log,
                       float* __restrict__ out,
                       int B)
{
    __shared__ float smem[ROWS_PER_BLK][32 * LDS_STRIDE];

    const int lane = threadIdx.x & 31;
    const int wave = threadIdx.x >> 5;
    const int row  = blockIdx.x * ROWS_PER_BLK + wave;
    if (row >= B) return;   // wave-uniform

    float* lds_row = &smem[wave][0];
    const uint32_t lds_base = (uint32_t)(uintptr_t)lds_row;   // low 32 bits = LDS byte offset
    const uint64_t xbase    = (uint64_t)(uintptr_t)x;
    const uint64_t obase    = (uint64_t)(uintptr_t)out;
    const uint32_t row_byte = (uint32_t)row * (uint32_t)(T_LEN * 4);

    // ---------------- Phase 0: TDM DMA row -> LDS, padding applied by hardware --------
    // D# group 0: count=1 | lds_addr | global_addr(57b) | type=2
    // D# group 1: data_size=4B, pad_enable, pad_interval=6 (128 DWORDs),
    //             pad_amount=3 (4 DWORDs)  -> effective LDS chunk stride = 132 floats.
    //             tensor_dim0=4096, tensor_dim1=1, tile=4096x1, dim0_stride=4096.
    {
        const uint64_t gaddr = xbase + (uint64_t)row_byte;
        v4u g0;
        g0.x = 1u;                                              // count=1 (valid), user mode
        g0.y = lds_base;                                        // lds_addr [63:32]
        g0.z = (u32)(gaddr & 0xFFFFFFFFu);                      // global_addr [95:64]
        g0.w = (u32)((gaddr >> 32) & 0x01FFFFFFu) | 0x80000000u; // addr[56:32] | type=2<<30
        v8i g1;
        g1[0] = (int)((2u << 16)      // data_size = 4 bytes
                    | (1u << 20)      // pad_enable
                    | (6u << 22)      // pad_interval: every 128 DWORDs
                    | (3u << 25));    // pad_amount: 4 DWORDs
        g1[1] = (int)(0x1000u << 16); // tensor_dim0 = 4096 (bits 63:48)
        g1[2] = (int)(1u << 16);      // tensor_dim1 = 1     (bits 95:80)
        g1[3] = (int)(0x1000u << 16); // tile_dim0 = 4096    (bits 127:112)
        g1[4] = 1;                    // tile_dim1 = 1
        g1[5] = 4096;                 // tensor_dim0_stride (low 32)
        g1[6] = 0;
        g1[7] = 0;
        const v4i gz = (v4i)0;        // groups 2/3 unused (2-D tensor)
#if defined(__clang_major__) && (__clang_major__ >= 23)
        const v8i gz8 = (v8i)0;
        __builtin_amdgcn_tensor_load_to_lds(g0, g1, gz, gz, gz8, 0);
#else
        __builtin_amdgcn_tensor_load_to_lds(g0, g1, gz, gz, 0);
#endif
    }

    // ---------------- Scalar parameters (overlap with the DMA) ----------------
    const float ea    = __expf(alpha_log[0]);
    const float eb    = __expf(beta_log[0]);
    const float omega = __expf(omega_log[0]);
    const float denom = 1.0f + ea + eb;
    const float alpha = ea / denom;
    const float beta  = eb / denom;
    float bp = beta;
#pragma unroll
    for (int q = 0; q < 7; ++q) bp *= bp;    // beta^128
    const float bp128 = bp;
    const float bp127 = bp128 / beta;        // beta > 0 strictly

    asm volatile("s_wait_tensorcnt 0x0" ::: "memory");  // row resident in LDS

    // ---------------- Pass A: per-lane chunk -> affine map (A,B) + variance partials --
    // Chunk i covers t in [128i, 128i+127]. Transfer: s_out = A * s_in + Bacc,
    // with s_in = s_{128i-1} (lane 0: s_in = s0 itself, A = beta^127).
    float sum = 0.0f, sumsq = 0.0f, Bacc = 0.0f;
    const int base_t = lane * CHUNK;
    float xprev = (lane > 0) ? lds_row[(lane - 1) * LDS_STRIDE + (CHUNK - 1)] : 0.0f;
#pragma unroll 4
    for (int m = 0; m < CHUNK; ++m) {
        const float xv = lds_row[lane * LDS_STRIDE + m];
        sum   += xv;
        sumsq += xv * xv;
        if (base_t + m > 0) {                           // false only for lane 0, m == 0
            const float c = fmaf(alpha, xprev * xprev, omega);
            Bacc = fmaf(beta, Bacc, c);
        }
        xprev = xv;
    }

    // ---------------- Wave reduction: unbiased variance (ddof = 1) ----------------
#pragma unroll
    for (int d = 16; d > 0; d >>= 1) {
        sum   += __shfl_xor(sum,   d, 32);
        sumsq += __shfl_xor(sumsq, d, 32);
    }
    const float Tf = (float)T_LEN;
    const float s0 = (sumsq - sum * sum / Tf) / (Tf - 1.0f);

    // ---------------- Wave inclusive scan over affine maps (5 shuffle steps) ---------
    float Asc = (lane == 0) ? bp127 : bp128;
    float Bsc = Bacc;
#pragma unroll
    for (int d = 1; d < 32; d <<= 1) {
        const float Ao = __shfl_up(Asc, d, 32);
        const float Bo = __shfl_up(Bsc, d, 32);
        if (lane >= d) {
            Bsc = fmaf(Asc, Bo, Bsc);   // compose: cur o prev
            Asc = Asc * Ao;
        }
    }
    // exclusive prefix -> entry state s_{128*lane - 1}
    const float Ae = __shfl_up(Asc, 1, 32);
    const float Be = __shfl_up(Bsc, 1, 32);
    float s = (lane == 0) ? s0 : fmaf(Ae, s0, Be);

    // ---------------- Pass B: materialize sqrt(sigma2) in place in LDS ---------------
    xprev = (lane > 0) ? lds_row[(lane - 1) * LDS_STRIDE + (CHUNK - 1)] : 0.0f;
#pragma unroll 4
    for (int m = 0; m < CHUNK; ++m) {
        const int t   = base_t + m;
        const int idx = lane * LDS_STRIDE + m;
        const float xcur = lds_row[idx];                // read slot BEFORE overwrite
        if (t > 0) {
            const float c = fmaf(alpha, xprev * xprev, omega);
            s = fmaf(beta, s, c);
        }
        lds_row[idx] = __builtin_sqrtf(s);
        xprev = xcur;
    }

    // ---------------- Phase C: async DMA LDS -> out (32 x 512B coalesced) ------------
    asm volatile("s_wait_dscnt 0x0" ::: "memory");      // LDS writes visible to DMA
    {
        uint32_t gv = row_byte + (uint32_t)lane * 16u;
        uint32_t lv = lds_base + (uint32_t)lane * 16u;
#pragma unroll
        for (int j = 0; j < 32; ++j) {
            asm volatile("global_store_async_from_lds_b128 %0, %1, %2"
                         :: "v"(gv), "v"(lv), "s"(obase) : "memory");
            gv += 512u;
            lv += (uint32_t)(LDS_STRIDE * 4);
        }
        asm volatile("s_wait_asynccnt 0x0" ::: "memory");
    }
}

extern "C" void kernel_launch(void* const* d_in, const int* in_sizes, int n_in,
                              void* d_out, int out_size, void* d_ws, size_t ws_size,
                              hipStream_t stream)
{
    const float* x         = (const float*)d_in[0];
    const float* omega_log = (const float*)d_in[1];
    const float* alpha_log = (const float*)d_in[2];
    const float* beta_log  = (const float*)d_in[3];
    float* out             = (float*)d_out;

    const int B = in_sizes[0] / T_LEN;
    const int grid = (B + ROWS_PER_BLK - 1) / ROWS_PER_BLK;

    garch_scan_kernel<<<grid, 32 * ROWS_PER_BLK, 0, stream>>>(
        x, omega_log, alpha_log, beta_log, out, B);
}